// SC_aggr_3221225472205
// MI455X (gfx1250) — compile-verified
//
#include <hip/hip_runtime.h>
#include <math.h>

typedef float v2f __attribute__((ext_vector_type(2)));
typedef float v4f __attribute__((ext_vector_type(4)));
typedef float v8f __attribute__((ext_vector_type(8)));
typedef int   i4  __attribute__((ext_vector_type(4)));

#define HID      128
#define HIDP     132   /* padded LDS row stride: 132 % 64 = 4 -> conflict-free column reads */
#define NCLS     8
#define NSCHEMES 5
#define KNEI     16

// ---------------------------------------------------------------------------
// Kernel 1: probs[n][:] = softmax( relu(feats[n] @ W1 + b1) @ W2 + b2 )
// One 16-node row tile per wave; hidden GEMM via V_WMMA_F32_16X16X4_F32.
//
// f32 WMMA fragment model (16x16x4, wave32):
//   A (16x4):  m = lane%16, khalf = lane/16; VGPR v holds A[m][k0 + 2*khalf + v]
//   B (4x16):  n = lane%16, khalf = lane/16; VGPR v holds B[k0 + 2*khalf + v][n]
//   C/D (16x16): VGPR r holds C[r + 8*(lane/16)][lane%16]
// ---------------------------------------------------------------------------
__launch_bounds__(128)
__global__ void mlp_probs_kernel(const float* __restrict__ feats,
                                 const float* __restrict__ W1,
                                 const float* __restrict__ b1,
                                 const float* __restrict__ W2,
                                 const float* __restrict__ b2,
                                 float* __restrict__ probs,
                                 int n_nodes)
{
    __shared__ float Hs[4][16][HIDP];   // ~33 KB, padded hidden tiles (one per wave)
    __shared__ float Ls[4][16][NCLS];   // 2 KB logits

    const int lane = threadIdx.x & 31;
    const int wave = threadIdx.x >> 5;
    const int m    = lane & 15;     // A row / B,C column within tile
    const int kh   = lane >> 4;     // K-half selector
    const int rowBase = (blockIdx.x * 4 + wave) * 16;

    int row = rowBase + m;
    if (row >= n_nodes) row = n_nodes - 1;          // clamp (cndmask, no EXEC change)
    const float* arow = feats + (size_t)row * HID;

    v8f zero = {};
    v8f acc[8];
#pragma unroll
    for (int j = 0; j < 8; ++j) acc[j] = zero;

    // H_tile[16x128] = A[16x128] @ W1[128x128], 32 K-tiles of 4, 8 column tiles
#pragma unroll 2
    for (int kt = 0; kt < 32; ++kt) {
        const int k0 = kt * 4 + 2 * kh;
        v2f a = *(const v2f*)(arow + k0);           // A[m][k0], A[m][k0+1]
#pragma unroll
        for (int j = 0; j < 8; ++j) {
            v2f b;
            b.x = W1[(size_t)(k0 + 0) * HID + 16 * j + m];
            b.y = W1[(size_t)(k0 + 1) * HID + 16 * j + m];
            acc[j] = __builtin_amdgcn_wmma_f32_16x16x4_f32(
                false, a, false, b, (short)0, acc[j], false, false);
        }
    }

    // bias + ReLU, stage hidden tile to LDS
#pragma unroll
    for (int j = 0; j < 8; ++j) {
        const int nn = 16 * j + m;
        const float bb = b1[nn];
#pragma unroll
        for (int r = 0; r < 8; ++r) {
            const int mm = r + 8 * kh;
            float h = acc[j][r] + bb;
            Hs[wave][mm][nn] = h > 0.0f ? h : 0.0f;
        }
    }
    __syncthreads();

    // logits[16x8] = H[16x128] @ W2[128x8]: each lane -> node m, classes 4*kh..4*kh+3
    const int cbase = 4 * kh;
    v4f lg = *(const v4f*)(b2 + cbase);
    for (int k = 0; k < HID; ++k) {
        const float h = Hs[wave][m][k];
        const v4f w = *(const v4f*)(W2 + k * NCLS + cbase);
        lg += h * w;
    }
    *(v4f*)(&Ls[wave][m][cbase]) = lg;
    __syncthreads();

    // softmax over 8 classes, lanes 0..15 each finish one node
    if (lane < 16) {
        const int n_out = rowBase + lane;
        if (n_out < n_nodes) {
            float v[NCLS];
            float mx = -3.0e38f;
#pragma unroll
            for (int c = 0; c < NCLS; ++c) { v[c] = Ls[wave][lane][c]; mx = fmaxf(mx, v[c]); }
            float s = 0.0f;
#pragma unroll
            for (int c = 0; c < NCLS; ++c) { v[c] = expf(v[c] - mx); s += v[c]; }
            const float inv = 1.0f / s;
#pragma unroll
            for (int c = 0; c < NCLS; ++c)
                probs[(size_t)n_out * NCLS + c] = v[c] * inv;
        }
    }
}

// ---------------------------------------------------------------------------
// Kernel 2: gather/aggregate. One thread per node.
//   scheme_pred = sum_s softmax(att)[s] * mean_k probs[sc_nei[s,n,k]]
//   out = sigmoid(alpha)*probs[n] + (1-sigmoid(alpha))*scheme_pred
// probs table (1.6 MB) is L2-resident; 80 independent 32B gathers per thread.
// ---------------------------------------------------------------------------
__launch_bounds__(256)
__global__ void aggregate_kernel(const int* __restrict__ sc_nei,
                                 const float* __restrict__ probs,
                                 const float* __restrict__ alpha,
                                 const float* __restrict__ attention,
                                 float* __restrict__ out,
                                 int n_nodes)
{
    const int n = blockIdx.x * blockDim.x + threadIdx.x;
    if (n >= n_nodes) return;

    // attention softmax over the 5 schemes
    float at[NSCHEMES];
    float mx = -3.0e38f;
#pragma unroll
    for (int s = 0; s < NSCHEMES; ++s) {
        at[s] = attention[(size_t)n * NSCHEMES + s];
        mx = fmaxf(mx, at[s]);
    }
    float asum = 0.0f;
#pragma unroll
    for (int s = 0; s < NSCHEMES; ++s) { at[s] = expf(at[s] - mx); asum += at[s]; }
    const float ainv = 1.0f / asum;

    v4f sp0 = {}, sp1 = {};
#pragma unroll
    for (int s = 0; s < NSCHEMES; ++s) {
        const i4* idx = (const i4*)(sc_nei + ((size_t)s * n_nodes + n) * KNEI);
        v4f t0 = {}, t1 = {};
#pragma unroll
        for (int q = 0; q < KNEI / 4; ++q) {
            const i4 id = idx[q];
#pragma unroll
            for (int e = 0; e < 4; ++e) {
                const v4f* p = (const v4f*)(probs + (size_t)id[e] * NCLS);
                t0 += p[0];
                t1 += p[1];
            }
        }
        const float w = at[s] * ainv * (1.0f / (float)KNEI);
        sp0 += w * t0;
        sp1 += w * t1;
    }

    const float g = 1.0f / (1.0f + expf(-alpha[n]));
    const v4f* sf = (const v4f*)(probs + (size_t)n * NCLS);
    v4f o0 = g * sf[0] + (1.0f - g) * sp0;
    v4f o1 = g * sf[1] + (1.0f - g) * sp1;
    *(v4f*)(out + (size_t)n * NCLS + 0) = o0;
    *(v4f*)(out + (size_t)n * NCLS + 4) = o1;
}

// ---------------------------------------------------------------------------
extern "C" void kernel_launch(void* const* d_in, const int* in_sizes, int n_in,
                              void* d_out, int out_size, void* d_ws, size_t ws_size,
                              hipStream_t stream)
{
    // setup_inputs order: sc_nei, feats, W1, b1, W2, b2, alpha, attention
    const int*   sc_nei    = (const int*)d_in[0];
    const float* feats     = (const float*)d_in[1];
    const float* W1        = (const float*)d_in[2];
    const float* b1        = (const float*)d_in[3];
    const float* W2        = (const float*)d_in[4];
    const float* b2        = (const float*)d_in[5];
    const float* alpha     = (const float*)d_in[6];
    const float* attention = (const float*)d_in[7];
    float* out = (float*)d_out;

    const int n_nodes = in_sizes[6];        // alpha is [N,1]
    float* probs = (float*)d_ws;            // N * 8 floats of scratch

    dim3 grid1((n_nodes + 63) / 64);        // 4 waves x 16 rows per block
    mlp_probs_kernel<<<grid1, 128, 0, stream>>>(feats, W1, b1, W2, b2, probs, n_nodes);

    dim3 grid2((n_nodes + 255) / 256);
    aggregate_kernel<<<grid2, 256, 0, stream>>>(sc_nei, probs, alpha, attention, out, n_nodes);
}